// MultiLevelHashEncoding_18872086298816
// MI455X (gfx1250) — compile-verified
//
#include <hip/hip_runtime.h>

// Multi-level hash-grid encoding (Instant-NGP style), MI455X / gfx1250.
//
// Strategy:
//  - Pre-pass: transpose emb (16,2,16384) -> (16,16384,2) into d_ws so each
//    trilinear corner gather is a single 8-byte LDS read (ds_load_b64).
//  - Main: grid.y = level (16), grid.x = point chunk (128), 256 threads.
//    Stage the level's full interleaved 128 KB table into LDS with CDNA5
//    async global->LDS b128 copies (immediate offsets, both LDS and global
//    addresses advance via INST_OFFSET), fence with s_wait_asynccnt.
//  - Each thread: trilinear weights with zero-padding validity folded per
//    dim, dense or XOR-prime hashed index, 8 float2 gathers from LDS, one
//    float2 (8B) store into the strided (B,16,2) output (L2 merges lines).

#define N_ENC      16384
#define N_ENC_MASK (N_ENC - 1)
#define NLEV       16
#define BPTS       (1u << 21)
#define THREADS    256
#define PCHUNKS    128
#define PTS_PER_BLOCK (BPTS / PCHUNKS)       // 16384
#define ITERS      (PTS_PER_BLOCK / THREADS) // 64

#define PRIME_Y 2654435761u
#define PRIME_Z 805459861u

__device__ __constant__ int c_res[NLEV] = {
    16, 20, 25, 32, 40, 50, 64, 80, 101, 128, 161, 203, 256, 322, 406, 512
};

// ---------------- pre-pass: interleave embedding planes ----------------
// emb: (16, 2, 16384) fp32  ->  embT: (16, 16384, 2) fp32
extern "C" __global__ __launch_bounds__(THREADS)
void emb_transpose_kernel(const float* __restrict__ emb,
                          float* __restrict__ embT)
{
    const unsigned i = blockIdx.x * THREADS + threadIdx.x;  // 0 .. 16*16384-1
    if (i < NLEV * N_ENC) {
        const unsigned l = i >> 14;          // / N_ENC
        const unsigned k = i & N_ENC_MASK;
        const float e0 = emb[(size_t)l * 2 * N_ENC + k];
        const float e1 = emb[(size_t)l * 2 * N_ENC + N_ENC + k];
        float2 v;
        v.x = e0;
        v.y = e1;
        *reinterpret_cast<float2*>(embT + (size_t)l * 2 * N_ENC + 2u * k) = v;
    }
}

// ---------------- main kernel ----------------
// INTER = true : tabsrc is interleaved (16,16384,2); corner gather = ds_load_b64
// INTER = false: tabsrc is two planes  (16,2,16384); corner gather = 2x ds_load_b32
template <bool INTER>
__global__ __launch_bounds__(THREADS, 2)
void hashenc_gfx1250_kernel(const float* __restrict__ x,
                            const float* __restrict__ tabsrc,
                            float* __restrict__ out)
{
    __shared__ float tab[2 * N_ENC];   // 131072 bytes (one level's table)

    const int   level = blockIdx.y;
    const int   res   = c_res[level];
    const float resf  = (float)res;
    const bool  dense = (res * res * res) <= N_ENC;

    // ---- async table fill: global -> LDS (flat 128 KB copy) ----
    {
        const char* g  = (const char*)(tabsrc + (size_t)level * (2 * N_ENC))
                         + threadIdx.x * 16;
        unsigned    lo = (unsigned)(size_t)(&tab[0]) + threadIdx.x * 16;
        // 131072 B / (256 threads * 16 B) = 32 passes; INST_OFFSET advances
        // both the LDS and global addresses per the async-to-LDS semantics.
#pragma unroll
        for (int it = 0; it < 32; ++it) {
            asm volatile("global_load_async_to_lds_b128 %0, %1, off offset:%c2"
                         :: "v"(lo), "v"(g), "i"(it * 4096) : "memory");
        }
        asm volatile("s_wait_asynccnt 0" ::: "memory");
    }
    __syncthreads();

    // ---- main point loop ----
    const size_t base_b = (size_t)blockIdx.x * PTS_PER_BLOCK + threadIdx.x;
    float* __restrict__ outp = out + level * 2;

    for (int it = 0; it < ITERS; ++it) {
        const size_t b = base_b + (size_t)it * THREADS;

        const float px = x[b * 3 + 0];
        const float py = x[b * 3 + 1];
        const float pz = x[b * 3 + 2];

        // grid_sample align_corners=False unnormalization
        const float ixf = ((px + 1.0f) * resf - 1.0f) * 0.5f;
        const float iyf = ((py + 1.0f) * resf - 1.0f) * 0.5f;
        const float izf = ((pz + 1.0f) * resf - 1.0f) * 0.5f;

        const float fx = floorf(ixf);
        const float fy = floorf(iyf);
        const float fz = floorf(izf);
        const float tx = ixf - fx;
        const float ty = iyf - fy;
        const float tz = izf - fz;

        const int ix0 = (int)fx, iy0 = (int)fy, iz0 = (int)fz;
        const int ix1 = ix0 + 1, iy1 = iy0 + 1, iz1 = iz0 + 1;

        // per-dim weights with zero-padding validity folded in
        float wx[2], wy[2], wz[2];
        wx[0] = (ix0 >= 0 && ix0 < res) ? (1.0f - tx) : 0.0f;
        wx[1] = (ix1 >= 0 && ix1 < res) ? tx          : 0.0f;
        wy[0] = (iy0 >= 0 && iy0 < res) ? (1.0f - ty) : 0.0f;
        wy[1] = (iy1 >= 0 && iy1 < res) ? ty          : 0.0f;
        wz[0] = (iz0 >= 0 && iz0 < res) ? (1.0f - tz) : 0.0f;
        wz[1] = (iz1 >= 0 && iz1 < res) ? tz          : 0.0f;

        // clamped corner coordinates
        const int rm1 = res - 1;
        unsigned cx[2], cy[2], cz[2];
        cx[0] = (unsigned)min(max(ix0, 0), rm1);
        cx[1] = (unsigned)min(max(ix1, 0), rm1);
        cy[0] = (unsigned)min(max(iy0, 0), rm1);
        cy[1] = (unsigned)min(max(iy1, 0), rm1);
        cz[0] = (unsigned)min(max(iz0, 0), rm1);
        cz[1] = (unsigned)min(max(iz1, 0), rm1);

        unsigned idx[8];
        if (dense) {
            const unsigned r  = (unsigned)res;
            const unsigned r2 = r * r;
            unsigned sx[2] = { cx[0],      cx[1]      };
            unsigned sy[2] = { cy[0] * r,  cy[1] * r  };
            unsigned sz[2] = { cz[0] * r2, cz[1] * r2 };
#pragma unroll
            for (int c = 0; c < 8; ++c)
                idx[c] = sx[c & 1] + sy[(c >> 1) & 1] + sz[c >> 2];
        } else {
            unsigned sx[2] = { cx[0],           cx[1]           };
            unsigned sy[2] = { cy[0] * PRIME_Y, cy[1] * PRIME_Y };
            unsigned sz[2] = { cz[0] * PRIME_Z, cz[1] * PRIME_Z };
#pragma unroll
            for (int c = 0; c < 8; ++c)
                idx[c] = (sx[c & 1] ^ sy[(c >> 1) & 1] ^ sz[c >> 2]) & N_ENC_MASK;
        }

        float f0 = 0.0f, f1 = 0.0f;
#pragma unroll
        for (int c = 0; c < 8; ++c) {
            const float w = wx[c & 1] * wy[(c >> 1) & 1] * wz[c >> 2];
            if (INTER) {
                const float2 e =
                    *reinterpret_cast<const float2*>(&tab[2u * idx[c]]);
                f0 = fmaf(e.x, w, f0);
                f1 = fmaf(e.y, w, f1);
            } else {
                f0 = fmaf(tab[idx[c]],         w, f0);
                f1 = fmaf(tab[N_ENC + idx[c]], w, f1);
            }
        }

        float2 r2v;
        r2v.x = f0;
        r2v.y = f1;
        *reinterpret_cast<float2*>(outp + b * 32) = r2v;   // out[b][level][0..1]
    }
}

extern "C" void kernel_launch(void* const* d_in, const int* in_sizes, int n_in,
                              void* d_out, int out_size, void* d_ws, size_t ws_size,
                              hipStream_t stream) {
    const float* x   = (const float*)d_in[0];   // (B, 3) fp32
    const float* emb = (const float*)d_in[1];   // (16, 2, 16384) fp32
    float*       out = (float*)d_out;           // (B, 16, 2) fp32

    const size_t need = (size_t)NLEV * 2 * N_ENC * sizeof(float);  // 2 MB
    dim3 grid(PCHUNKS, NLEV, 1);
    dim3 block(THREADS, 1, 1);

    if (ws_size >= need && d_ws != nullptr) {
        float* embT = (float*)d_ws;
        const unsigned total  = NLEV * N_ENC;
        const unsigned blocks = (total + THREADS - 1) / THREADS;
        hipLaunchKernelGGL(emb_transpose_kernel, dim3(blocks), block, 0, stream,
                           emb, embT);
        hipLaunchKernelGGL(HIP_KERNEL_NAME(hashenc_gfx1250_kernel<true>),
                           grid, block, 0, stream, x, embT, out);
    } else {
        hipLaunchKernelGGL(HIP_KERNEL_NAME(hashenc_gfx1250_kernel<false>),
                           grid, block, 0, stream, x, emb, out);
    }
}